// SkeletonBase_34445637714245
// MI455X (gfx1250) — compile-verified
//
#include <hip/hip_runtime.h>

// ---------------------------------------------------------------------------
// Batched H36M forward kinematics, memory-bound streaming kernel for gfx1250.
// 302MB read + 96MB write @ 23.3 TB/s -> ~17us floor; compute is negligible.
// Data movement uses CDNA5 async global->LDS copies (ASYNCcnt) so each wave
// streams its 32 poses with fully coalesced 512B-per-instruction transfers,
// double-buffered in 4-joint stages, instead of 1152B-strided per-lane loads.
// ---------------------------------------------------------------------------

#define POSE_BYTES      1152            // 32 joints * 9 f32
#define STAGE_JOINTS    4
#define STAGE_BYTES     144             // 4 joints * 36B
#define LDS_POSE_STRIDE 176             // 144B chunk + 32B pad (16B aligned, 2-way bank conflict max)
#define WAVE_POSES      32
#define WAVES_PER_BLOCK 4
#define BLOCK_THREADS   (WAVES_PER_BLOCK * 32)
#define WAVE_BUF        (WAVE_POSES * LDS_POSE_STRIDE)   // 5632 B per stage buffer
#define WAVE_LDS        (2 * WAVE_BUF)                   // double buffered
#define BLOCK_LDS       (WAVES_PER_BLOCK * WAVE_LDS)     // 45056 B (< 64KB static limit)

namespace {
constexpr int kPar[32] = {-1, 0, 1, 2, 3, 4, 0, 6, 7, 8, 9, 0, 11, 12, 13, 14,
                          12, 16, 17, 18, 19, 20, 19, 22, 12, 24, 25, 26, 27, 28, 27, 30};
}

// Process one joint: G[j] = X @ G[parent]; P[j] = offs[j] @ G[parent] + P[parent].
// All indices are compile-time constants after unrolling -> arrays are SROA'd
// into registers; the compiler keeps only the live subtree (~4 matrices).
#define FK_JOINT(j, X)                                                                     \
  do {                                                                                     \
    if ((j) == 0) {                                                                        \
      _Pragma("unroll") for (int t = 0; t < 9; ++t) G[0][t] = (X)[t];                      \
      P[0][0] = 0.f; P[0][1] = 0.f; P[0][2] = 0.f;                                         \
    } else {                                                                               \
      const int pp = kPar[(j)];                                                            \
      const float o0 = offs[(j) * 3 + 0];                                                  \
      const float o1 = offs[(j) * 3 + 1];                                                  \
      const float o2 = offs[(j) * 3 + 2];                                                  \
      _Pragma("unroll") for (int l = 0; l < 3; ++l) {                                      \
        G[(j)][0 + l] = fmaf((X)[0], G[pp][l], fmaf((X)[1], G[pp][3 + l], (X)[2] * G[pp][6 + l])); \
        G[(j)][3 + l] = fmaf((X)[3], G[pp][l], fmaf((X)[4], G[pp][3 + l], (X)[5] * G[pp][6 + l])); \
        G[(j)][6 + l] = fmaf((X)[6], G[pp][l], fmaf((X)[7], G[pp][3 + l], (X)[8] * G[pp][6 + l])); \
        P[(j)][l] = fmaf(o0, G[pp][l], fmaf(o1, G[pp][3 + l], fmaf(o2, G[pp][6 + l], P[pp][l])));  \
      }                                                                                    \
    }                                                                                      \
    op[(j) * 3 + 0] = P[(j)][0];                                                           \
    op[(j) * 3 + 1] = P[(j)][1];                                                           \
    op[(j) * 3 + 2] = P[(j)][2];                                                           \
  } while (0)

__global__ __launch_bounds__(BLOCK_THREADS) void fk_async_kernel(
    const float* __restrict__ x, const float* __restrict__ offs,
    float* __restrict__ out, int n) {
  __shared__ char smem[BLOCK_LDS];

  const int lane = threadIdx.x & 31;
  const int wave = threadIdx.x >> 5;
  const long long waveBase = (long long)blockIdx.x * BLOCK_THREADS + wave * 32;
  const long long pose = waveBase + lane;

  float G[32][9];
  float P[32][3];

  // Staged path requires: whole wave in-bounds, and byte offsets fit in the
  // 32-bit VADDR of the GVS-mode async load (true for N = 262144 -> 302MB).
  const bool staged = (waveBase + WAVE_POSES <= (long long)n) &&
                      ((long long)n * POSE_BYTES < (1LL << 31));

  if (staged) {
    // Low 32 bits of a generic shared pointer == LDS byte offset (aperture is
    // 4GiB aligned; addrspacecast AS3->flat is {aperture_hi, offset}).
    const unsigned lds0 = (unsigned)(unsigned long long)(smem + wave * WAVE_LDS);
    const unsigned gbase = (unsigned)(waveBase * POSE_BYTES);

    // One stage = 4 joints x 32 poses = 4608B, moved by 9 coalesced async
    // b128 instructions (lane L copies 16B at chunk offset i*512 + L*16).
    auto issue_stage = [&](int s, int b) {
      const unsigned lb = lds0 + (unsigned)(b * WAVE_BUF);
#pragma unroll
      for (int i = 0; i < 9; ++i) {
        const int e = i * 32 + lane;      // 16B element index within stage
        const int p = e / 9;              // pose within wave chunk
        const int r = e - p * 9;          // 16B unit within 144B joint chunk
        const unsigned goff = gbase + (unsigned)(p * POSE_BYTES + s * STAGE_BYTES + r * 16);
        const unsigned loff = lb + (unsigned)(p * LDS_POSE_STRIDE + r * 16);
        asm volatile("global_load_async_to_lds_b128 %0, %1, %2"
                     :: "v"(loff), "v"(goff), "s"(x)
                     : "memory");
      }
    };

    float* op = out + pose * 96LL;
    issue_stage(0, 0);
#pragma unroll
    for (int s = 0; s < 8; ++s) {
      if (s < 7) {
        // WAR guard: stage s-1 reads of buffer (s+1)&1 must retire before the
        // async engine overwrites it.
        asm volatile("s_wait_dscnt 0x0" ::: "memory");
        issue_stage(s + 1, (s + 1) & 1);
        // Async loads retire in order: <=9 outstanding => stage s complete.
        asm volatile("s_wait_asynccnt 0x9" ::: "memory");
      } else {
        asm volatile("s_wait_asynccnt 0x0" ::: "memory");
      }
      const float* cb = (const float*)(smem + wave * WAVE_LDS + (s & 1) * WAVE_BUF +
                                       lane * LDS_POSE_STRIDE);
#pragma unroll
      for (int jj = 0; jj < STAGE_JOINTS; ++jj) {
        const int j = s * STAGE_JOINTS + jj;
        float X[9];
#pragma unroll
        for (int t = 0; t < 9; ++t) X[t] = cb[jj * 9 + t];
        FK_JOINT(j, X);
      }
    }
  } else if (pose < (long long)n) {
    // Tail / fallback: direct global loads.
    const float* xp = x + pose * 288LL;
    float* op = out + pose * 96LL;
#pragma unroll
    for (int j = 0; j < 32; ++j) {
      float X[9];
#pragma unroll
      for (int t = 0; t < 9; ++t) X[t] = xp[j * 9 + t];
      FK_JOINT(j, X);
    }
  }
}

extern "C" void kernel_launch(void* const* d_in, const int* in_sizes, int n_in,
                              void* d_out, int out_size, void* d_ws, size_t ws_size,
                              hipStream_t stream) {
  (void)n_in; (void)out_size; (void)d_ws; (void)ws_size;
  const float* x    = (const float*)d_in[0];   // (N, 32, 3, 3) f32
  const float* offs = (const float*)d_in[1];   // (32, 3) f32
  float* out        = (float*)d_out;           // (N, 32, 3) f32

  const int n = in_sizes[0] / 288;             // poses
  const int blocks = (n + BLOCK_THREADS - 1) / BLOCK_THREADS;
  fk_async_kernel<<<blocks, BLOCK_THREADS, 0, stream>>>(x, offs, out, n);
}